// FocalLoss_90443421319419
// MI455X (gfx1250) — compile-verified
//
#include <hip/hip_runtime.h>
#include <hip/hip_bf16.h>

// ---------------------------------------------------------------------------
// RetinaNet focal loss on MI455X (gfx1250, wave32).
// Memory-bound: 168MB classification stream dominates -> fused single pass,
// float4 coalesced loads, LDS-resident GT boxes, WMMA-based wave reductions.
// ---------------------------------------------------------------------------

typedef __attribute__((ext_vector_type(2))) float v2f;
typedef __attribute__((ext_vector_type(8))) float v8f;

#if defined(__has_builtin)
#  if __has_builtin(__builtin_amdgcn_wmma_f32_16x16x4_f32)
#    define USE_WMMA_RED 1
#  endif
#endif
#ifndef USE_WMMA_RED
#  define USE_WMMA_RED 0
#endif

#define NUM_CLASSES 80
#define BLOCK 256
#define MAX_M 64

// Wave32 sum-reduction. With B == all-ones, D[m,n] = sum_k A[m,k].
// A layout (16x4 f32): lane L<16 -> A[L, {0,1}] = {v0,v1}; lane L>=16 ->
// A[L-16, {2,3}]. We put the value in v0 and 0 in v1, so rowsum[m] =
// val[m] + val[m+16]. Each lane sums its 8 D VGPRs:
//   lanes 0..15  get sum(rows 0..7), lanes 16..31 get sum(rows 8..15),
// and one xor-16 shuffle completes the 32-lane total (valid in all lanes).
__device__ __forceinline__ float wave_reduce_sum(float val) {
#if USE_WMMA_RED
    v2f a; a.x = val; a.y = 0.0f;
    v2f bones; bones.x = 1.0f; bones.y = 1.0f;
    v8f c = {};
    c = __builtin_amdgcn_wmma_f32_16x16x4_f32(
        /*neg_a=*/false, a, /*neg_b=*/false, bones,
        /*c_mod=*/(short)0, c, /*reuse_a=*/false, /*reuse_b=*/false);
    float s = c[0] + c[1] + c[2] + c[3] + c[4] + c[5] + c[6] + c[7];
    s += __shfl_xor(s, 16, 32);
    return s;
#else
    for (int off = 16; off > 0; off >>= 1) val += __shfl_xor(val, off, 32);
    return val;
#endif
}

__device__ __forceinline__ float smooth_l1(float d) {
    d = fabsf(d);
    return (d <= (1.0f / 9.0f)) ? (4.5f * d * d) : (d - (0.5f / 9.0f));
}

// One focal-loss element. meta: bits[7:0]=label, bit8=pos, bit9=neg(clear).
__device__ __forceinline__ float focal_elem(float x, int cls_idx, int meta) {
    bool pos = (meta & 256) != 0;
    bool neg = (meta & 512) != 0;
    float c = fminf(fmaxf(x, 1e-4f), 1.0f - 1e-4f);
    bool is_one = pos && (cls_idx == (meta & 255));
    float fw  = is_one ? (1.0f - c) : c;
    float bce = -__logf(is_one ? c : (1.0f - c));
    float w = (pos || neg) ? 0.5f : 0.0f;   // ALPHA=0.5 -> constant alpha factor
    return w * fw * fw * bce;               // GAMMA=2 -> square
}

__global__ void __launch_bounds__(BLOCK)
focal_main_kernel(const float* __restrict__ cls,      // [B,A,C]
                  const float* __restrict__ reg,      // [B,A,4]
                  const float* __restrict__ anchors,  // [A,4]
                  const float* __restrict__ ann,      // [B,M,5]
                  float* __restrict__ ws,             // [B,3] cls/reg/npos
                  int A, int M) {
    __shared__ float s_ann[MAX_M * 5];
    __shared__ int   s_meta[BLOCK];
    __shared__ float s_wred[BLOCK / 32][3];

    const int tid = threadIdx.x;
    const int b   = blockIdx.y;
    const int a0  = blockIdx.x * BLOCK;
    const int a   = a0 + tid;

    for (int i = tid; i < M * 5; i += BLOCK)
        s_ann[i] = ann[(size_t)b * M * 5 + i];
    __syncthreads();

    // ---- Phase 1: IoU assignment + regression loss (one anchor per thread)
    const float4 ab = ((const float4*)anchors)[a];
    const float aw  = ab.z - ab.x;
    const float ah  = ab.w - ab.y;
    const float acx = ab.x + 0.5f * aw;
    const float acy = ab.y + 0.5f * ah;
    const float area_a = aw * ah;

    float best = -1e30f;
    int   barg = 0;
    for (int m = 0; m < M; ++m) {
        const float* g = &s_ann[m * 5];
        const float bx1 = g[0], by1 = g[1], bx2 = g[2], by2 = g[3];
        const float area_b = (bx2 - bx1) * (by2 - by1);
        float iw = fminf(ab.z, bx2) - fmaxf(ab.x, bx1);
        float ih = fminf(ab.w, by2) - fmaxf(ab.y, by1);
        iw = fmaxf(iw, 0.0f);
        ih = fmaxf(ih, 0.0f);
        const float inter = iw * ih;
        const float ua    = fmaxf(area_a + area_b - inter, 1e-8f);
        const float iou   = inter / ua;
        if (iou > best) { best = iou; barg = m; }  // first-max (argmax semantics)
    }

    const bool pos = best >= 0.5f;
    const bool neg = best < 0.4f;
    const int  label = (int)s_ann[barg * 5 + 4];
    s_meta[tid] = label | (pos ? 256 : 0) | (neg ? 512 : 0);

    float reg_partial = 0.0f;
    if (pos) {
        const float* g = &s_ann[barg * 5];
        float gw = g[2] - g[0], gh = g[3] - g[1];
        const float gcx = g[0] + 0.5f * gw;
        const float gcy = g[1] + 0.5f * gh;
        gw = fmaxf(gw, 1.0f);
        gh = fmaxf(gh, 1.0f);
        const float t0 = (gcx - acx) / aw * 10.0f;   // /0.1
        const float t1 = (gcy - acy) / ah * 10.0f;   // /0.1
        const float t2 = __logf(gw / aw) * 5.0f;     // /0.2
        const float t3 = __logf(gh / ah) * 5.0f;     // /0.2
        const float4 r = ((const float4*)reg)[(size_t)b * A + a];
        reg_partial = smooth_l1(t0 - r.x) + smooth_l1(t1 - r.y) +
                      smooth_l1(t2 - r.z) + smooth_l1(t3 - r.w);
    }
    __syncthreads();

    // ---- Phase 2: focal loss sweep, lane-coalesced float4 stream.
    // Block covers BLOCK anchors * 80 classes = 5120 float4s; C=80 % 4 == 0
    // so every float4 stays inside one anchor's class row.
    const float4* cp = (const float4*)(cls + ((size_t)b * A + a0) * NUM_CLASSES);
    float cls_partial = 0.0f;
#pragma unroll
    for (int k = 0; k < (BLOCK * NUM_CLASSES) / (4 * BLOCK); ++k) {  // 20 iters
        const int v  = tid + k * BLOCK;
        const int al = v / (NUM_CLASSES / 4);                 // v / 20
        const int c0 = (v - al * (NUM_CLASSES / 4)) * 4;
        const int meta = s_meta[al];
        const float4 x = cp[v];
        cls_partial += focal_elem(x.x, c0 + 0, meta);
        cls_partial += focal_elem(x.y, c0 + 1, meta);
        cls_partial += focal_elem(x.z, c0 + 2, meta);
        cls_partial += focal_elem(x.w, c0 + 3, meta);
    }

    // ---- Reductions: v_wmma_f32_16x16x4_f32 wave sums; ballot for pos count.
    const float wc = wave_reduce_sum(cls_partial);
    const float wr = wave_reduce_sum(reg_partial);
    const float wp = (float)__popcll(__ballot(pos));

    const int wave = tid >> 5;
    if ((tid & 31) == 0) {
        s_wred[wave][0] = wc;
        s_wred[wave][1] = wr;
        s_wred[wave][2] = wp;
    }
    __syncthreads();
    if (tid == 0) {
        float tc = 0.0f, tr = 0.0f, tp = 0.0f;
        for (int w = 0; w < BLOCK / 32; ++w) {
            tc += s_wred[w][0];
            tr += s_wred[w][1];
            tp += s_wred[w][2];
        }
        atomicAdd(&ws[b * 3 + 0], tc);
        atomicAdd(&ws[b * 3 + 1], tr);
        atomicAdd(&ws[b * 3 + 2], tp);
    }
}

__global__ void focal_init_kernel(float* ws, int n) {
    int i = blockIdx.x * blockDim.x + threadIdx.x;
    if (i < n) ws[i] = 0.0f;
}

__global__ void focal_finalize_kernel(const float* __restrict__ ws,
                                      float* __restrict__ out, int B) {
    if (threadIdx.x == 0 && blockIdx.x == 0) {
        float cm = 0.0f, rm = 0.0f;
        for (int b = 0; b < B; ++b) {
            const float cs = ws[b * 3 + 0];
            const float rs = ws[b * 3 + 1];
            const float np = ws[b * 3 + 2];
            cm += cs / fmaxf(np, 1.0f);
            rm += (np > 0.0f) ? (rs / fmaxf(np * 4.0f, 1.0f)) : 0.0f;
        }
        out[0] = cm / (float)B;
        out[1] = rm / (float)B;
    }
}

extern "C" void kernel_launch(void* const* d_in, const int* in_sizes, int n_in,
                              void* d_out, int out_size, void* d_ws, size_t ws_size,
                              hipStream_t stream) {
    (void)n_in; (void)out_size; (void)ws_size;
    const float* cls     = (const float*)d_in[0];  // [B,A,80]
    const float* reg     = (const float*)d_in[1];  // [B,A,4]
    const float* anchors = (const float*)d_in[2];  // [1,A,4]
    const float* ann     = (const float*)d_in[3];  // [B,M,5]
    float* out = (float*)d_out;
    float* ws  = (float*)d_ws;

    const int A = in_sizes[2] / 4;
    const int B = in_sizes[1] / (A * 4);
    const int M = in_sizes[3] / (B * 5);

    focal_init_kernel<<<1, 32, 0, stream>>>(ws, B * 3);
    dim3 grid(A / BLOCK, B);
    focal_main_kernel<<<grid, BLOCK, 0, stream>>>(cls, reg, anchors, ann, ws, A, M);
    focal_finalize_kernel<<<1, 32, 0, stream>>>(ws, out, B);
}